// Fature_embedding_47347719471684
// MI455X (gfx1250) — compile-verified
//
#include <hip/hip_runtime.h>

// FFM feature layer: B=16384 rows, F=16 fields, V=300000 vocab, D=10 dims.
// Out row = [120 pairs x 10 | 16 self x 10 | 16 linear] = 1376 floats.
//
// Memory-bound random gather: emb table is 192 MB == MI455X L2 capacity.
// Strategy: per-row scatter-gather of 256 embedding vectors into LDS via
// gfx1250 async loads (ASYNCcnt path), compute Hadamard pair products from
// LDS, stream results out with non-temporal stores so the 90 MB output
// stream does not evict the emb table from L2.

#define FDIM   16
#define DDIM   10
#define NPAIR  120
#define PAIR_ELEMS (NPAIR * DDIM)   // 1200
#define SELF_ELEMS (FDIM * DDIM)    // 160
#define ROW_OUT    (PAIR_ELEMS + SELF_ELEMS + FDIM) // 1376
#define WAVES_PER_BLOCK 4
#define BLOCK_THREADS   (WAVES_PER_BLOCK * 32)
#define VECS    (FDIM * FDIM)       // 256 gathered vectors per row
#define ROW_LDS (VECS * DDIM)       // 2560 floats per row staged in LDS

#ifndef HAVE_ASYNC
#if defined(__has_builtin)
#if __has_builtin(__builtin_amdgcn_global_load_async_to_lds_b64)
#define HAVE_ASYNC 1
#endif
#endif
#endif
#ifndef HAVE_ASYNC
#define HAVE_ASYNC 0
#endif

#ifndef HAVE_WAIT_ASYNC
#if defined(__has_builtin)
#if __has_builtin(__builtin_amdgcn_s_wait_asynccnt)
#define HAVE_WAIT_ASYNC 1
#endif
#endif
#endif
#ifndef HAVE_WAIT_ASYNC
#define HAVE_WAIT_ASYNC 0
#endif

// b64 async payload type per clang diagnostic: int __attribute__((vector_size(8)))
typedef int v2i32 __attribute__((vector_size(8)));
typedef __attribute__((address_space(1))) v2i32 g_v2i32_t;
typedef __attribute__((address_space(3))) v2i32 l_v2i32_t;

// generic->AS3: AMDGPU flat LDS addresses carry the LDS byte offset in the
// low 32 bits (ISA 10.2), so truncation is the cast.
static __device__ __forceinline__ l_v2i32_t* as_lds_v2(void* p) {
  return (l_v2i32_t*)(unsigned)(unsigned long long)p;
}
static __device__ __forceinline__ g_v2i32_t* as_global_v2(const void* p) {
  return (g_v2i32_t*)(unsigned long long)p;
}

static __device__ __forceinline__ void wait_async_zero() {
#if HAVE_WAIT_ASYNC
  __builtin_amdgcn_s_wait_asynccnt(0);
#else
  asm volatile("s_wait_asynccnt 0" ::: "memory");
#endif
}

__global__ __launch_bounds__(BLOCK_THREADS)
void ffm_gather_kernel(const int* __restrict__ x,
                       const float* __restrict__ emb,
                       const float* __restrict__ lin,
                       float* __restrict__ out,
                       int B, long long V) {
  __shared__ float sE[WAVES_PER_BLOCK * ROW_LDS];  // 40 KB
  __shared__ int   sPair[NPAIR];

  const int tid  = threadIdx.x;
  const int lane = tid & 31;
  const int wave = tid >> 5;

  // Build triu_indices(16, k=1) pair table once per block (row-major order).
  if (tid < NPAIR) {
    int p = tid, i = 0, rem = p;
    while (rem >= (FDIM - 1 - i)) { rem -= (FDIM - 1 - i); ++i; }
    sPair[tid] = (i << 8) | (i + 1 + rem);
  }

  const int b = blockIdx.x * WAVES_PER_BLOCK + wave;  // one row per wave
  float* Ew = &sE[wave * ROW_LDS];

  int idx_own = 0;
  if (b < B) {
    if (lane < FDIM) idx_own = x[b * FDIM + lane];

    // Gather 256 vectors E[g][f][0..9] = emb[g, x[b,f], :].
    // Vector id t = g*16 + f; lane handles t = k*32 + lane, k = 0..7.
    // Each vector is 40 B at 8-B alignment -> 5x async b64 per vector.
#pragma unroll
    for (int k = 0; k < 8; ++k) {
      const int t = k * 32 + lane;
      const int f = t & 15;
      const int g = t >> 4;
      const int vidx = __shfl(idx_own, f, 32);
      const float* src = emb + ((long long)g * V + (long long)vidx) * DDIM;
      float* dst = Ew + t * DDIM;
#if HAVE_ASYNC
      g_v2i32_t* gs = as_global_v2(src);
      l_v2i32_t* ls = as_lds_v2(dst);
      __builtin_amdgcn_global_load_async_to_lds_b64(gs, ls, 0, 0);
      __builtin_amdgcn_global_load_async_to_lds_b64(gs, ls, 8, 0);
      __builtin_amdgcn_global_load_async_to_lds_b64(gs, ls, 16, 0);
      __builtin_amdgcn_global_load_async_to_lds_b64(gs, ls, 24, 0);
      __builtin_amdgcn_global_load_async_to_lds_b64(gs, ls, 32, 0);
#else
      const float2* s2 = (const float2*)src;
      float2*       d2 = (float2*)dst;
#pragma unroll
      for (int q = 0; q < 5; ++q) d2[q] = s2[q];
#endif
    }
  }

#if HAVE_ASYNC
  wait_async_zero();   // retire this wave's async LDS writes
#endif
  __syncthreads();     // pair table + cross-lane LDS visibility

  if (b >= B) return;

  float* orow = out + (long long)b * ROW_OUT;

  // Pair term: out[e] = E[j][i][d] * E[i][j][d], e = p*10 + d.
  for (int e = lane; e < PAIR_ELEMS; e += 32) {
    const int p  = (int)((unsigned)e / DDIM);
    const int d  = e - p * DDIM;
    const int ij = sPair[p];
    const int i  = ij >> 8;
    const int j  = ij & 255;
    const float va = Ew[(j * FDIM + i) * DDIM + d];  // emb[j, x[b,i], d]
    const float vb = Ew[(i * FDIM + j) * DDIM + d];  // emb[i, x[b,j], d]
    __builtin_nontemporal_store(va * vb, &orow[e]);
  }

  // Self embeddings: emb[f, x[b,f], d].
  for (int e = lane; e < SELF_ELEMS; e += 32) {
    const int f = (int)((unsigned)e / DDIM);
    const int d = e - f * DDIM;
    __builtin_nontemporal_store(Ew[(f * FDIM + f) * DDIM + d],
                                &orow[PAIR_ELEMS + e]);
  }

  // Linear term: lin[x[b,f]].
  if (lane < FDIM) {
    __builtin_nontemporal_store(lin[idx_own],
                                &orow[PAIR_ELEMS + SELF_ELEMS + lane]);
  }
}

extern "C" void kernel_launch(void* const* d_in, const int* in_sizes, int n_in,
                              void* d_out, int out_size, void* d_ws, size_t ws_size,
                              hipStream_t stream) {
  const int*   x   = (const int*)d_in[0];
  const float* emb = (const float*)d_in[1];
  const float* lin = (const float*)d_in[2];
  float*       out = (float*)d_out;

  const int B = in_sizes[0] / FDIM;
  const long long V = (long long)in_sizes[1] / (FDIM * DDIM);

  const int blocks = (B + WAVES_PER_BLOCK - 1) / WAVES_PER_BLOCK;
  ffm_gather_kernel<<<blocks, BLOCK_THREADS, 0, stream>>>(x, emb, lin, out, B, V);
}